// RelModel0_5162550690000
// MI455X (gfx1250) — compile-verified
//
#include <hip/hip_runtime.h>
#include <hip/hip_bf16.h>
#include <stddef.h>

// MI455X / gfx1250 (CDNA5, wave32, WMMA) implementation.
//
// out = relu(concat(U_l @ Wl^T, U_r @ Wr^T)) @ rel_W
//   U_l/U_r: [4096, 2058] built on the fly from slice differences of f/b + type embs
//   Wl/Wr  : head_W or dep_W per `form`    rel_W: [2048, 6]
//
// fp32 end-to-end via V_WMMA_F32_16X16X4_F32 (matches reference dtype).
// Weights (16.8 MB) are L2-resident (192 MB L2): B fragments stream from L2 as
// per-lane b64 loads (K-contiguous in a weight row). The 32-row A tile (diff
// features) is staged chunk-wise in LDS and shared by all 8 waves; each B
// fragment feeds TWO WMMAs (two 16-row A tiles) -> 16 FLOP per L2 byte.

typedef float v2f __attribute__((ext_vector_type(2)));
typedef float v4f __attribute__((ext_vector_type(4)));
typedef float v8f __attribute__((ext_vector_type(8)));

#define BB   4096
#define SS   64
#define HH   1024
#define TD   10
#define RR   6
#define KIN  2058          // 2*H + TD
#define KCHUNK 420         // 4*105 (odd*4 -> conflict-free row banks); 5*420 = 2100 >= 2058
#define NCHUNKS 5
#define MTILE 32           // rows per block (two 16-row WMMA tiles)

__global__ __launch_bounds__(256)
void rel_gemm_relu_kernel(const float* __restrict__ f,
                          const float* __restrict__ b,
                          const float* __restrict__ lte,
                          const float* __restrict__ rte,
                          const float* __restrict__ headW,
                          const float* __restrict__ depW,
                          float* __restrict__ Cws,        // [4096, 2048] relu'd
                          const int* __restrict__ ps, const int* __restrict__ pe,
                          const int* __restrict__ pc0, const int* __restrict__ pc1,
                          const int* __restrict__ pf)
{
    __shared__ float ldsA[MTILE * KCHUNK];   // 53.8 KB: one K-chunk of 32-row A tile

    const int tid    = threadIdx.x;
    const int wave   = tid >> 5;          // 8 waves (wave32)
    const int lane   = tid & 31;
    const int laneHi = lane >> 4;         // 0 | 1 (selects K pair / M half)
    const int laneLo = lane & 15;         // N column within 16-wide tile

    const int start = *ps, end = *pe, cut0 = *pc0, cut1 = *pc1, form = *pf;

    const int side  = blockIdx.y >> 1;    // 0 = left operand, 1 = right operand
    const int nhalf = blockIdx.y & 1;     // which 512-wide half of N=1024
    const int br0   = blockIdx.x * MTILE; // batch-row tile base

    int sf0, sf1, sb0, sb1;
    const float* tp;
    const float* W;
    if (side == 0) {
        sf0 = cut0;      sf1 = start - 1;
        sb0 = start - 1; sb1 = cut0;
        tp = lte;
        W  = (form == 2) ? depW : headW;  // form 0/1 -> head, 2 -> dep
    } else {
        sf0 = end;       sf1 = cut1 - 1;
        sb0 = cut1 - 1;  sb1 = end;
        tp = rte;
        W  = (form == 1) ? depW : headW;  // form 0/2 -> head, 1 -> dep
    }

    // ---- per-thread staging setup: fixed row, float4 columns ----
    const int frow = tid >> 3;            // 0..31 : LDS row this thread fills
    const int fcol = tid & 7;             // float4 column phase within row
    const int grow = br0 + frow;          // global batch row
    const float* fp0 = f + ((size_t)grow * SS + sf0) * HH;
    const float* fp1 = f + ((size_t)grow * SS + sf1) * HH;
    const float* bp0 = b + ((size_t)grow * SS + sb0) * HH;
    const float* bp1 = b + ((size_t)grow * SS + sb1) * HH;
    const float* tpp = tp + grow * TD;
    float* ldsRow = ldsA + frow * KCHUNK;

    // ---- per-wave compute setup ----
    const int nb = nhalf * 512 + wave * 64;   // this wave's N base (64 columns)

    // Weight-row pointers for the 4 N-tiles (W is [1024, 2058] row-major, so
    // B[k, n] = W[n*2058 + k]; K contiguous -> one b64 load per B fragment).
    const float* wrow0 = W + (size_t)(nb +  0 + laneLo) * KIN;
    const float* wrow1 = W + (size_t)(nb + 16 + laneLo) * KIN;
    const float* wrow2 = W + (size_t)(nb + 32 + laneLo) * KIN;
    const float* wrow3 = W + (size_t)(nb + 48 + laneLo) * KIN;

    v8f accA0 = {}, accA1 = {}, accA2 = {}, accA3 = {};   // rows br0 +  0..15
    v8f accB0 = {}, accB1 = {}, accB2 = {}, accB3 = {};   // rows br0 + 16..31

    for (int kc = 0; kc < NCHUNKS; ++kc) {
        const int kbase = kc * KCHUNK;

        __syncthreads();   // previous chunk's compute done before overwrite
        // ---- stage A chunk into LDS (vectorized, no div/mod) ----
        for (int c4 = fcol; c4 < KCHUNK / 4; c4 += 8) {
            const int gc = kbase + 4 * c4;   // global K column (multiple of 4)
            v4f v;
            if (gc < HH) {
                v = *(const v4f*)(fp0 + gc) - *(const v4f*)(fp1 + gc);
            } else if (gc < 2 * HH) {
                v = *(const v4f*)(bp0 + (gc - HH)) - *(const v4f*)(bp1 + (gc - HH));
            } else {
#pragma unroll
                for (int j = 0; j < 4; ++j)
                    v[j] = (gc + j < KIN) ? tpp[gc + j - 2 * HH] : 0.0f;
            }
            *(v4f*)(ldsRow + 4 * c4) = v;
        }
        __syncthreads();

        // ---- compute: 105 K-steps, 8 x V_WMMA_F32_16X16X4_F32 per step ----
        for (int kk = 0; kk < KCHUNK; kk += 4) {
            // A fragments (16x4 f32): lanes 0-15 hold row=laneLo, K=kk..kk+1;
            // lanes 16-31 hold K=kk+2..kk+3 (ISA 32-bit A-matrix layout).
            const v2f a0 = *(const v2f*)&ldsA[laneLo * KCHUNK + kk + 2 * laneHi];
            const v2f a1 = *(const v2f*)&ldsA[(16 + laneLo) * KCHUNK + kk + 2 * laneHi];

            // B fragment (4x16 f32): VGPR r, lane l -> K = base + r + 2*(l>=16),
            // N = l&15. Clamp the K tail in-bounds (A is zero there, 0*x = 0).
            int kg = kbase + kk + 2 * laneHi;
            if (kg > KIN - 2) kg = KIN - 2;

            const v2f b0 = *(const v2f*)(wrow0 + kg);
            const v2f b1 = *(const v2f*)(wrow1 + kg);
            const v2f b2 = *(const v2f*)(wrow2 + kg);
            const v2f b3 = *(const v2f*)(wrow3 + kg);

            accA0 = __builtin_amdgcn_wmma_f32_16x16x4_f32(false, a0, false, b0,
                                                          (short)0, accA0, false, false);
            accB0 = __builtin_amdgcn_wmma_f32_16x16x4_f32(false, a1, false, b0,
                                                          (short)0, accB0, false, false);
            accA1 = __builtin_amdgcn_wmma_f32_16x16x4_f32(false, a0, false, b1,
                                                          (short)0, accA1, false, false);
            accB1 = __builtin_amdgcn_wmma_f32_16x16x4_f32(false, a1, false, b1,
                                                          (short)0, accB1, false, false);
            accA2 = __builtin_amdgcn_wmma_f32_16x16x4_f32(false, a0, false, b2,
                                                          (short)0, accA2, false, false);
            accB2 = __builtin_amdgcn_wmma_f32_16x16x4_f32(false, a1, false, b2,
                                                          (short)0, accB2, false, false);
            accA3 = __builtin_amdgcn_wmma_f32_16x16x4_f32(false, a0, false, b3,
                                                          (short)0, accA3, false, false);
            accB3 = __builtin_amdgcn_wmma_f32_16x16x4_f32(false, a1, false, b3,
                                                          (short)0, accB3, false, false);
        }
    }

    // ---- epilogue: fused ReLU, store to workspace U = [4096, 2048] ----
    // C/D layout: VGPR v, lane l -> M = v + 8*(l>=16), N = l&15.
    const size_t colBase = (size_t)side * HH + nb + laneLo;
#pragma unroll
    for (int v = 0; v < 8; ++v) {
        const int m = v + 8 * laneHi;
        float* o0 = Cws + (size_t)(br0 + m) * (2 * HH) + colBase;
        o0[ 0] = fmaxf(accA0[v], 0.0f);
        o0[16] = fmaxf(accA1[v], 0.0f);
        o0[32] = fmaxf(accA2[v], 0.0f);
        o0[48] = fmaxf(accA3[v], 0.0f);
        float* o1 = Cws + (size_t)(br0 + 16 + m) * (2 * HH) + colBase;
        o1[ 0] = fmaxf(accB0[v], 0.0f);
        o1[16] = fmaxf(accB1[v], 0.0f);
        o1[32] = fmaxf(accB2[v], 0.0f);
        o1[48] = fmaxf(accB3[v], 0.0f);
    }
}

// out[b, r] = sum_n U[b, n] * rel_W[n, r]   (U already relu'd; rel_W: [2048, 6])
__global__ __launch_bounds__(256)
void rel_reduce_kernel(const float* __restrict__ U,
                       const float* __restrict__ relW,
                       float* __restrict__ out)
{
    __shared__ float red[RR * 256];
    const int row = blockIdx.x;
    const int tid = threadIdx.x;

    float acc[RR] = {0.f, 0.f, 0.f, 0.f, 0.f, 0.f};
    const float* urow = U + (size_t)row * (2 * HH);
    for (int n = tid; n < 2 * HH; n += 256) {
        const float u = urow[n];
        const float* wr = relW + n * RR;
#pragma unroll
        for (int r = 0; r < RR; ++r) acc[r] += u * wr[r];
    }
#pragma unroll
    for (int r = 0; r < RR; ++r) red[r * 256 + tid] = acc[r];
    __syncthreads();
    for (int s = 128; s > 0; s >>= 1) {
        if (tid < s) {
#pragma unroll
            for (int r = 0; r < RR; ++r)
                red[r * 256 + tid] += red[r * 256 + tid + s];
        }
        __syncthreads();
    }
    if (tid < RR) out[(size_t)row * RR + tid] = red[tid * 256];
}

extern "C" void kernel_launch(void* const* d_in, const int* in_sizes, int n_in,
                              void* d_out, int out_size, void* d_ws, size_t ws_size,
                              hipStream_t stream) {
    const float* f     = (const float*)d_in[0];
    const float* b     = (const float*)d_in[1];
    const float* lte   = (const float*)d_in[2];
    const float* rte   = (const float*)d_in[3];
    const float* headW = (const float*)d_in[4];
    const float* depW  = (const float*)d_in[5];
    const float* relW  = (const float*)d_in[6];
    const int*   ps    = (const int*)d_in[7];
    const int*   pe    = (const int*)d_in[8];
    const int*   pc0   = (const int*)d_in[9];
    const int*   pc1   = (const int*)d_in[10];
    const int*   pf    = (const int*)d_in[11];

    float* Cws = (float*)d_ws;            // 4096 * 2048 * 4 B = 33.5 MB

    // grid.x = batch tiles (4096/32), grid.y = {left,right} x {N half}
    dim3 grid1(BB / MTILE, 4, 1);
    rel_gemm_relu_kernel<<<grid1, 256, 0, stream>>>(
        f, b, lte, rte, headW, depW, Cws, ps, pe, pc0, pc1, pf);

    rel_reduce_kernel<<<BB, 256, 0, stream>>>(Cws, relW, (float*)d_out);
}